// GCNGraphClassifier_36928128811252
// MI455X (gfx1250) — compile-verified
//
#include <hip/hip_runtime.h>

typedef __attribute__((ext_vector_type(2))) float v2f;
typedef __attribute__((ext_vector_type(8))) float v8f;

#define N_NODES 50000
#define N_EDGES 640000

// ---------------------------------------------------------------------------
// C[M,N] = A[M,K] @ W[K,N] using V_WMMA_F32_16X16X4_F32 (full fp32 matrix pipe)
// One wave computes one 16x16 tile; block covers 16 rows x N cols.
// A tile (16 x K) is staged once per block in LDS (padded stride to kill bank
// conflicts); each wave reads its A fragments from LDS via ds_load.
// M multiple of 16 (50000 = 3125*16), N multiple of 16, K multiple of 4.
// ---------------------------------------------------------------------------
template <int K>
__global__ __launch_bounds__(256) void k_gemm_wmma(const float* __restrict__ A,
                                                   const float* __restrict__ W,
                                                   float* __restrict__ C,
                                                   int N) {
  constexpr int LDA = K + 4;             // 132: stride % 64 banks == 4 -> conflict-free
  __shared__ float sA[16 * LDA];

  const int lane = threadIdx.x & 31;
  const int wave = threadIdx.x >> 5;
  const int m0   = blockIdx.x << 4;
  const int n0   = wave << 4;
  const int l15  = lane & 15;
  const int koff = (lane >> 4) << 1;     // 0 or 2

  // Stage A tile: 16 rows x K floats, coalesced float4 global loads.
  const float4* gA = (const float4*)(A + (size_t)m0 * K);
  for (int i = threadIdx.x; i < 16 * (K / 4); i += blockDim.x) {
    int r = i / (K / 4);
    int c = i % (K / 4);
    *(float4*)&sA[r * LDA + (c << 2)] = gA[(size_t)r * (K / 4) + c];
  }
  __syncthreads();

  // A fragment: lane holds row (l15), K = k + koff + {0,1} (8B-aligned pair in LDS)
  const float* arow = sA + l15 * LDA;
  // B fragment: lane holds col (l15), K = k + koff + {0,1}
  const float* wcol = W + n0 + l15;

  v8f acc = {};
#pragma unroll 4
  for (int k = 0; k < K; k += 4) {
    v2f a, b;
    a.x = arow[k + koff];
    a.y = arow[k + koff + 1];
    b.x = wcol[(size_t)(k + koff) * N];
    b.y = wcol[(size_t)(k + koff + 1) * N];
    acc = __builtin_amdgcn_wmma_f32_16x16x4_f32(
        /*neg_a=*/false, a, /*neg_b=*/false, b,
        /*c_mod=*/(short)0, acc, /*reuse_a=*/false, /*reuse_b=*/false);
  }

  // C/D layout: lane L, vgpr v -> (M = v + 8*(L>>4), N = L&15)
  float* crow = C + (size_t)(m0 + ((lane >> 4) << 3)) * N + n0 + l15;
#pragma unroll
  for (int v = 0; v < 8; ++v) crow[(size_t)v * N] = acc[v];
}

// ---------------------------------------------------------------------------
// Degree: deg[col[e]] += 1 over edges (self-loop +1 folded into rsqrt kernel)
// ---------------------------------------------------------------------------
__global__ void k_degree(const long long* __restrict__ col, float* __restrict__ deg) {
  int e = blockIdx.x * blockDim.x + threadIdx.x;
  if (e < N_EDGES) atomicAdd(&deg[(int)col[e]], 1.0f);
}

__global__ void k_rsqrt_deg(float* __restrict__ deg) {
  int n = blockIdx.x * blockDim.x + threadIdx.x;
  if (n < N_NODES) deg[n] = rsqrtf(deg[n] + 1.0f);  // +1 for self-loop; always > 0
}

// ---------------------------------------------------------------------------
// Edge scatter: agg[col] += h[row] * dis[row]*dis[col], float4 per thread.
// C channels, (C/4) threads per edge.
// ---------------------------------------------------------------------------
template <int C, int LOG_TPE>
__global__ void k_scatter(const long long* __restrict__ row,
                          const long long* __restrict__ col,
                          const float* __restrict__ dis,
                          const float* __restrict__ h,
                          float* __restrict__ agg) {
  long long tid = (long long)blockIdx.x * blockDim.x + threadIdx.x;
  int e = (int)(tid >> LOG_TPE);
  if (e >= N_EDGES) return;
  int c = (int)(tid & ((1 << LOG_TPE) - 1)) << 2;
  int r = (int)row[e];
  int d = (int)col[e];
  float nrm = dis[r] * dis[d];
  float4 v = *(const float4*)(h + (size_t)r * C + c);
  float* dst = agg + (size_t)d * C + c;
  atomicAdd(dst + 0, v.x * nrm);
  atomicAdd(dst + 1, v.y * nrm);
  atomicAdd(dst + 2, v.z * nrm);
  atomicAdd(dst + 3, v.w * nrm);
}

// ---------------------------------------------------------------------------
// Finalize: out += h * dis^2 (self-loop message) + bias, optional ReLU.
// ---------------------------------------------------------------------------
template <int LOGC, bool RELU>
__global__ void k_finalize(const float* __restrict__ h,
                           const float* __restrict__ dis,
                           const float* __restrict__ bias,
                           float* __restrict__ out) {
  long long i = (long long)blockIdx.x * blockDim.x + threadIdx.x;
  if (i >= ((long long)N_NODES << LOGC)) return;
  int n = (int)(i >> LOGC);
  int c = (int)(i & ((1 << LOGC) - 1));
  float ds = dis[n];
  float v = out[i] + h[i] * ds * ds + bias[c];
  out[i] = RELU ? fmaxf(v, 0.0f) : v;
}

extern "C" void kernel_launch(void* const* d_in, const int* in_sizes, int n_in,
                              void* d_out, int out_size, void* d_ws, size_t ws_size,
                              hipStream_t stream) {
  const float*     x   = (const float*)d_in[0];
  const long long* ei  = (const long long*)d_in[1];   // int64 [2, E]
  const float*     W1  = (const float*)d_in[2];
  const float*     b1  = (const float*)d_in[3];
  const float*     W2  = (const float*)d_in[4];
  const float*     b2  = (const float*)d_in[5];
  const long long* row = ei;             // source nodes
  const long long* col = ei + N_EDGES;   // destination nodes

  // Workspace layout (floats):
  float* dis  = (float*)d_ws;                      // N_NODES (deg, then deg^-1/2)
  float* h1   = dis + 50176;                       // N_NODES*128  (16B aligned)
  float* agg1 = h1 + (size_t)N_NODES * 128;        // N_NODES*128
  float* h2   = agg1 + (size_t)N_NODES * 128;      // N_NODES*64
  float* out  = (float*)d_out;                     // N_NODES*64

  hipMemsetAsync(dis, 0, (size_t)N_NODES * sizeof(float), stream);
  hipMemsetAsync(agg1, 0, (size_t)N_NODES * 128 * sizeof(float), stream);
  hipMemsetAsync(out, 0, (size_t)N_NODES * 64 * sizeof(float), stream);

  // Degrees -> deg^{-1/2}
  k_degree<<<(N_EDGES + 255) / 256, 256, 0, stream>>>(col, dis);
  k_rsqrt_deg<<<(N_NODES + 255) / 256, 256, 0, stream>>>(dis);

  // ---- Layer 1 ----
  // h1 = x @ W1   (N=128 -> 8 waves/block)
  k_gemm_wmma<128><<<N_NODES / 16, 256, 0, stream>>>(x, W1, h1, 128);

  long long t1 = (long long)N_EDGES * 32;  // 128/4 threads per edge
  k_scatter<128, 5><<<(int)((t1 + 255) / 256), 256, 0, stream>>>(row, col, dis, h1, agg1);

  long long f1 = (long long)N_NODES * 128;
  k_finalize<7, true><<<(int)((f1 + 255) / 256), 256, 0, stream>>>(h1, dis, b1, agg1);

  // ---- Layer 2 ----
  // h2 = relu_out @ W2   (N=64 -> 4 waves/block)
  k_gemm_wmma<128><<<N_NODES / 16, 128, 0, stream>>>(agg1, W2, h2, 64);

  long long t2 = (long long)N_EDGES * 16;  // 64/4 threads per edge
  k_scatter<64, 4><<<(int)((t2 + 255) / 256), 256, 0, stream>>>(row, col, dis, h2, out);

  long long f2 = (long long)N_NODES * 64;
  k_finalize<6, false><<<(int)((f2 + 255) / 256), 256, 0, stream>>>(h2, dis, b2, out);
}